// CMCScore_triplet_11244224381540
// MI455X (gfx1250) — compile-verified
//
#include <hip/hip_runtime.h>

typedef float v2f __attribute__((ext_vector_type(2)));
typedef float v4f __attribute__((ext_vector_type(4)));
typedef float v8f __attribute__((ext_vector_type(8)));

#define NB    128        // batch B
#define DD    128        // feature dim D
#define KP1   4097       // K+1 candidates per query
#define KMAX  4096       // max valid k index
#define NN    1000000    // memory rows N
#define TILE_PITCH 132   // 128 + 4 floats pad -> conflict-free LDS reads

__device__ __forceinline__ float wave_sum(float v) {
#pragma unroll
  for (int o = 16; o > 0; o >>= 1) v += __shfl_xor(v, o, 32);
  return v;
}

// ---------------------------------------------------------------------------
// Phase 1+2: per (query b, direction pair) compute relu(1 - dis + pos).
// dis^2 for 16 candidates at once = diag( A x (A - 2f)^T ) + ||f||^2,
// accumulated with v_wmma_f32_16x16x4_f32 over D=128 in K=4 chunks.
// ---------------------------------------------------------------------------
__global__ void __launch_bounds__(128)
score_kernel(const float* __restrict__ l, const float* __restrict__ ab,
             const int* __restrict__ idx,
             const float* __restrict__ memory_l, const float* __restrict__ memory_ab,
             float* __restrict__ out) {
  __shared__ __align__(16) float sF[DD];
  __shared__ float sMeta[2];                       // [0]=||f||^2  [1]=pos
  __shared__ __align__(16) float sTile[4][16 * TILE_PITCH];

  const int b    = blockIdx.x;
  const int pair = blockIdx.y;                     // 0: out_l (mem_ab, l) ; 1: out_ab (mem_l, ab)
  const int lane = threadIdx.x & 31;
  const int wave = threadIdx.x >> 5;

  const float* x    = (pair == 0 ? l : ab) + b * DD;
  const float* mem  = (pair == 0 ? memory_ab : memory_l);
  const int*   idxb = idx + b * KP1;
  float* outp = out + (size_t)pair * ((size_t)NB * KP1) + (size_t)b * KP1;

  if (wave == 0) {
    // normalize feature: f = x / (||x|| + eps)
    v4f x4 = ((const v4f*)x)[lane];
    float s = wave_sum(x4.x * x4.x + x4.y * x4.y + x4.z * x4.z + x4.w * x4.w);
    float inv = 1.0f / (sqrtf(s) + 1e-7f);
    v4f f4 = x4 * inv;
    ((v4f*)sF)[lane] = f4;
    float ff = wave_sum(f4.x * f4.x + f4.y * f4.y + f4.z * f4.z + f4.w * f4.w);
    // pos = distance to candidate k=0
    const float* row0 = mem + (size_t)idxb[0] * DD;
    v4f w4 = ((const v4f*)row0)[lane];
    v4f d4 = w4 - f4;
    float ps = wave_sum(d4.x * d4.x + d4.y * d4.y + d4.z * d4.z + d4.w * d4.w);
    if (lane == 0) { sMeta[0] = ff; sMeta[1] = sqrtf(fmaxf(ps, 0.0f)); }
  }
  __syncthreads();

  const float ff  = sMeta[0];
  const float pos = sMeta[1];
  const int h = lane >> 4;                         // which K-pair this lane feeds
  const int r = lane & 15;                         // A-matrix row owned by this lane
  float* myTile = &sTile[wave][0];

  for (int tile = blockIdx.z * 4 + wave; tile <= KMAX / 16; tile += gridDim.z * 4) {
    // stage 16 gathered rows (512B each, fully coalesced) into padded LDS
    for (int p = 0; p < 16; ++p) {
      int k = tile * 16 + p;
      if (k > KMAX) k = KMAX;                      // clamp tail tile
      const size_t ri = (size_t)idxb[k];
      v4f w4 = *(const v4f*)(mem + ri * DD + (size_t)(lane * 4));
      *(v4f*)(myTile + p * TILE_PITCH + lane * 4) = w4;
    }

    v8f acc0 = {0.f, 0.f, 0.f, 0.f, 0.f, 0.f, 0.f, 0.f};
    v8f acc1 = {0.f, 0.f, 0.f, 0.f, 0.f, 0.f, 0.f, 0.f};
#pragma unroll
    for (int t = 0; t < 32; t += 2) {
      {
        int off = 4 * t + 2 * h;
        v2f a  = *(const v2f*)(myTile + r * TILE_PITCH + off);
        v2f fc = *(const v2f*)(sF + off);
        v2f bb = a - 2.0f * fc;                    // B = A - 2f, identical lane layout
        acc0 = __builtin_amdgcn_wmma_f32_16x16x4_f32(false, a, false, bb,
                                                     (short)0, acc0, false, false);
      }
      {
        int off = 4 * (t + 1) + 2 * h;
        v2f a  = *(const v2f*)(myTile + r * TILE_PITCH + off);
        v2f fc = *(const v2f*)(sF + off);
        v2f bb = a - 2.0f * fc;
        acc1 = __builtin_amdgcn_wmma_f32_16x16x4_f32(false, a, false, bb,
                                                     (short)0, acc1, false, false);
      }
    }
    v8f acc = acc0 + acc1;

    // diagonal extraction: rows 0-7 on lanes 0-7 (vgpr=lane),
    // rows 8-15 on lanes 24-31 (vgpr=lane-24)
    const bool have = (lane < 8) || (lane >= 24);
    const int  vi   = (lane < 8) ? lane : (lane - 24);
    float g = acc[0];
#pragma unroll
    for (int v = 1; v < 8; ++v) g = (vi == v) ? acc[v] : g;

    const int row = (lane < 8) ? lane : (lane - 16);
    const int k = tile * 16 + row;
    if (have && k <= KMAX) {
      float dis = sqrtf(fmaxf(g + ff, 0.0f));      // ||w||^2 - 2<w,f> + ||f||^2
      float sc = 1.0f - dis + pos;
      outp[k] = sc > 0.0f ? sc : 0.0f;
    }
  }
}

// ---------------------------------------------------------------------------
// Phase 3a: stream both 512MB memory tables to the output with NT hints
// (pure bandwidth; keep it out of L2).
// ---------------------------------------------------------------------------
__global__ void __launch_bounds__(256)
copy_kernel(const float* __restrict__ s0, const float* __restrict__ s1,
            float* __restrict__ d0, float* __restrict__ d1) {
  const v4f* s = (const v4f*)(blockIdx.y == 0 ? s0 : s1);
  v4f*       d = (v4f*)(blockIdx.y == 0 ? d0 : d1);
  const size_t n4 = (size_t)NN * DD / 4;
  const size_t stride = (size_t)gridDim.x * blockDim.x;
  for (size_t i = (size_t)blockIdx.x * blockDim.x + threadIdx.x; i < n4; i += stride) {
    v4f v = __builtin_nontemporal_load(s + i);
    __builtin_nontemporal_store(v, d + i);
  }
}

// ---------------------------------------------------------------------------
// Phase 3b: rewrite the 128 rows selected by y (momentum + renormalize).
// One wave per table, serial over b -> deterministic last-write-wins on
// duplicate y (matches JAX scatter semantics).
// ---------------------------------------------------------------------------
__global__ void __launch_bounds__(32)
update_kernel(const float* __restrict__ l, const float* __restrict__ ab,
              const int* __restrict__ y,
              const float* __restrict__ memory_l, const float* __restrict__ memory_ab,
              float* __restrict__ out) {
  const int which = blockIdx.x;                    // 0: memory_l/l ; 1: memory_ab/ab
  const int lane = threadIdx.x;
  const float* xsrc = (which == 0) ? l : ab;
  const float* mem  = (which == 0) ? memory_l : memory_ab;
  float* dst = out + (size_t)2 * NB * KP1 + (size_t)which * ((size_t)NN * DD);

  for (int b = 0; b < NB; ++b) {
    const size_t yy = (size_t)y[b];
    v4f x4 = ((const v4f*)(xsrc + b * DD))[lane];
    float s = wave_sum(x4.x * x4.x + x4.y * x4.y + x4.z * x4.z + x4.w * x4.w);
    v4f xn = x4 * (1.0f / (sqrtf(s) + 1e-7f));
    v4f m4 = *(const v4f*)(mem + yy * DD + (size_t)(lane * 4));
    v4f p4 = m4 * 0.5f + xn * 0.5f;                // momentum = 0.5
    float s2 = wave_sum(p4.x * p4.x + p4.y * p4.y + p4.z * p4.z + p4.w * p4.w);
    v4f o4 = p4 * (1.0f / sqrtf(s2));              // reference: no eps here
    *(v4f*)(dst + yy * DD + (size_t)(lane * 4)) = o4;
  }
}

extern "C" void kernel_launch(void* const* d_in, const int* in_sizes, int n_in,
                              void* d_out, int out_size, void* d_ws, size_t ws_size,
                              hipStream_t stream) {
  const float* l   = (const float*)d_in[0];
  const float* ab  = (const float*)d_in[1];
  const int*   y   = (const int*)d_in[2];
  const int*   idx = (const int*)d_in[3];
  const float* ml  = (const float*)d_in[4];
  const float* mab = (const float*)d_in[5];
  float* out = (float*)d_out;

  (void)in_sizes; (void)n_in; (void)out_size; (void)d_ws; (void)ws_size;

  // scores (gather + WMMA Gram-diagonal distances)
  score_kernel<<<dim3(NB, 2, 16), 128, 0, stream>>>(l, ab, idx, ml, mab, out);

  // memory table copy (dominant 2GB stream)
  float* dml  = out + (size_t)2 * NB * KP1;
  float* dmab = dml + (size_t)NN * DD;
  copy_kernel<<<dim3(4096, 2, 1), 256, 0, stream>>>(ml, mab, dml, dmab);

  // scatter-update the 128 selected rows (after the copy, same stream)
  update_kernel<<<dim3(2), 32, 0, stream>>>(l, ab, y, ml, mab, out);
}